// CombinedModel_29953101922630
// MI455X (gfx1250) — compile-verified
//
#include <hip/hip_runtime.h>

#define NN 50000           // N_NODES
#define NE 800000          // N_EDGES
#define B_ 16384           // BATCH
#define KDNN 256           // DNN_IN
#define F_ 64              // feature width (DNN_H == GNN_IN == GNN_H == MID)
#define BN_EPS 1e-5f

typedef __attribute__((ext_vector_type(2))) float v2f;
typedef __attribute__((ext_vector_type(8))) float v8f;

__device__ __forceinline__ void atomAddF(float* p, float v) { unsafeAtomicAdd(p, v); }

// ---------------------------------------------------------------------------
// WMMA fp32 GEMM: C[M,64] = A[M,K] @ W[K,64] (+ bias). M % 16 == 0, K % 4 == 0.
// One wave computes a 16x64 strip: 4 accumulators, 4*K/4 v_wmma_f32_16x16x4_f32.
// A-matrix lane layout (ISA 7.12.2, 32-bit A 16x4): lane m (0..15) holds row m,
// K={0,1}; lane m+16 holds row m, K={2,3}. B symmetric; C/D standard 8-VGPR.
// ---------------------------------------------------------------------------
__global__ void gemm_wmma_f32_n64(const float* __restrict__ A,
                                  const float* __restrict__ W,
                                  const float* __restrict__ bias,
                                  float* __restrict__ C, int M, int K) {
  const int lane = threadIdx.x & 31;
  const int wave = threadIdx.x >> 5;
  const int tile = blockIdx.x * (blockDim.x >> 5) + wave;
  const int row0 = tile * 16;
  if (row0 >= M) return;                    // wave-uniform exit, EXEC stays all-1s

  const int m  = lane & 15;
  const int kb = (lane >> 4) << 1;          // 0 for lanes 0-15, 2 for lanes 16-31
  const int n  = lane & 15;

  v8f acc0 = {}, acc1 = {}, acc2 = {}, acc3 = {};
  const float* Arow = A + (size_t)(row0 + m) * K;

  for (int k = 0; k < K; k += 4) {
    v2f a;  a.x = Arow[k + kb];  a.y = Arow[k + kb + 1];
    const float* Wr0 = W + (size_t)(k + kb) * F_;
    const float* Wr1 = Wr0 + F_;
    v2f b0; b0.x = Wr0[n];      b0.y = Wr1[n];
    v2f b1; b1.x = Wr0[16 + n]; b1.y = Wr1[16 + n];
    v2f b2; b2.x = Wr0[32 + n]; b2.y = Wr1[32 + n];
    v2f b3; b3.x = Wr0[48 + n]; b3.y = Wr1[48 + n];
    acc0 = __builtin_amdgcn_wmma_f32_16x16x4_f32(false, a, false, b0, (short)0, acc0, false, false);
    acc1 = __builtin_amdgcn_wmma_f32_16x16x4_f32(false, a, false, b1, (short)0, acc1, false, false);
    acc2 = __builtin_amdgcn_wmma_f32_16x16x4_f32(false, a, false, b2, (short)0, acc2, false, false);
    acc3 = __builtin_amdgcn_wmma_f32_16x16x4_f32(false, a, false, b3, (short)0, acc3, false, false);
  }

  float bb0 = bias ? bias[n]      : 0.f;
  float bb1 = bias ? bias[16 + n] : 0.f;
  float bb2 = bias ? bias[32 + n] : 0.f;
  float bb3 = bias ? bias[48 + n] : 0.f;
  const int rbase = row0 + ((lane >> 4) << 3);   // +8 for upper half-wave
#pragma unroll
  for (int r = 0; r < 8; ++r) {
    float* Crow = C + (size_t)(rbase + r) * F_;
    Crow[n]      = acc0[r] + bb0;
    Crow[16 + n] = acc1[r] + bb1;
    Crow[32 + n] = acc2[r] + bb2;
    Crow[48 + n] = acc3[r] + bb3;
  }
}

// --------------------------- GCN support kernels ---------------------------
__global__ void k_init(float* __restrict__ agg, float* __restrict__ deg,
                       float* __restrict__ gacc) {
  int t = blockIdx.x * blockDim.x + threadIdx.x;
  if (t < NN * F_) agg[t] = 0.f;
  if (t < NN)      deg[t] = 1.0f;     // self-loop
  if (t < F_)      gacc[t] = 0.f;
}

__global__ void k_zero(float* __restrict__ p, int n) {
  int t = blockIdx.x * blockDim.x + threadIdx.x;
  if (t < n) p[t] = 0.f;
}

__global__ void k_degree(const long long* __restrict__ ei, float* __restrict__ deg) {
  int e = blockIdx.x * blockDim.x + threadIdx.x;
  if (e >= NE) return;
  atomAddF(&deg[(int)ei[NE + e]], 1.0f);
}

__global__ void k_rsqrt_inplace(float* __restrict__ d, int n) {
  int t = blockIdx.x * blockDim.x + threadIdx.x;
  if (t < n) d[t] = rsqrtf(d[t]);
}

// scatter: agg[col] += dis[row]*dis[col] * H[row], 4 feats per thread
__global__ void k_aggregate(const long long* __restrict__ ei,
                            const float* __restrict__ dis,
                            const float* __restrict__ H,
                            float* __restrict__ agg) {
  long long t = (long long)blockIdx.x * blockDim.x + threadIdx.x;
  if (t >= (long long)NE * 16) return;
  int e = (int)(t >> 4);
  int q = ((int)t & 15) << 2;
  int r = (int)ei[e];
  int c = (int)ei[NE + e];
  float norm = dis[r] * dis[c];
  float4 hv = *(const float4*)(H + (size_t)r * F_ + q);
  float* dst = agg + (size_t)c * F_ + q;
  atomAddF(dst + 0, norm * hv.x);
  atomAddF(dst + 1, norm * hv.y);
  atomAddF(dst + 2, norm * hv.z);
  atomAddF(dst + 3, norm * hv.w);
}

// g = relu(agg + dis^2 * H + b), in place over agg
__global__ void k_combine_relu(float* __restrict__ agg, const float* __restrict__ H,
                               const float* __restrict__ dis, const float* __restrict__ b) {
  int t = blockIdx.x * blockDim.x + threadIdx.x;
  if (t >= NN * F_) return;
  int i = t >> 6, f = t & 63;
  float d = dis[i];
  float v = agg[t] + d * d * H[t] + b[f];
  agg[t] = v > 0.f ? v : 0.f;
}

// column sums of (agg + dis^2 * H) over all nodes -> gacc[64] (bias folded later)
__global__ void k_combine_mean(const float* __restrict__ agg, const float* __restrict__ H,
                               const float* __restrict__ dis, float* __restrict__ gacc) {
  __shared__ float sd[256];
  int f = threadIdx.x & 63;
  int g = threadIdx.x >> 6;                       // 0..3 node sub-groups
  float p = 0.f;
  for (int i = blockIdx.x * 4 + g; i < NN; i += gridDim.x * 4) {
    float d = dis[i];
    size_t idx = (size_t)i * F_ + f;
    p += agg[idx] + d * d * H[idx];
  }
  sd[threadIdx.x] = p;
  __syncthreads();
  if (threadIdx.x < 64)
    atomAddF(&gacc[threadIdx.x],
             sd[threadIdx.x] + sd[threadIdx.x + 64] + sd[threadIdx.x + 128] + sd[threadIdx.x + 192]);
}

// ----------------------------- DNN branch ---------------------------------
__global__ void k_bnstats(const float* __restrict__ H, float* __restrict__ mu,
                          float* __restrict__ istd) {
  __shared__ float s1[256], s2[256];
  int c = blockIdx.x;
  float a = 0.f, b = 0.f;
  for (int i = threadIdx.x; i < B_; i += 256) {
    float v = H[(size_t)i * F_ + c];
    a += v; b += v * v;
  }
  s1[threadIdx.x] = a; s2[threadIdx.x] = b;
  __syncthreads();
  for (int s = 128; s > 0; s >>= 1) {
    if (threadIdx.x < s) { s1[threadIdx.x] += s1[threadIdx.x + s]; s2[threadIdx.x] += s2[threadIdx.x + s]; }
    __syncthreads();
  }
  if (threadIdx.x == 0) {
    float m = s1[0] / (float)B_;
    float var = s2[0] / (float)B_ - m * m;     // population variance (ddof=0)
    mu[c] = m;
    istd[c] = rsqrtf(var + BN_EPS);
  }
}

__global__ void k_bnapply_relu(float* __restrict__ H, const float* __restrict__ mu,
                               const float* __restrict__ istd, const float* __restrict__ gam,
                               const float* __restrict__ bet) {
  int t = blockIdx.x * blockDim.x + threadIdx.x;
  if (t >= B_ * F_) return;
  int f = t & 63;
  float v = (H[t] - mu[f]) * istd[f] * gam[f] + bet[f];
  H[t] = v > 0.f ? v : 0.f;
}

// --------------------------- folded MLP tail -------------------------------
// wc[k] = sum_j fc1_W[k][j]*fc2_W[j]; bcomb = fc2_b + sum_j fc1_b[j]*fc2_W[j]
__global__ void k_foldw(const float* __restrict__ fc1W, const float* __restrict__ fc1b,
                        const float* __restrict__ fc2W, const float* __restrict__ fc2b,
                        float* __restrict__ wc, float* __restrict__ bcomb) {
  int k = threadIdx.x;                     // 128 threads
  float s = 0.f;
  for (int j = 0; j < F_; ++j) s += fc1W[k * F_ + j] * fc2W[j];
  wc[k] = s;
  if (k == 0) {
    float bb = fc2b[0];
    for (int j = 0; j < F_; ++j) bb += fc1b[j] * fc2W[j];
    *bcomb = bb;
  }
}

// sval = bcomb + dot(gacc/N + gcn2_b, wc[64..127])
__global__ void k_gnn_scalar(const float* __restrict__ gacc, const float* __restrict__ g2b,
                             const float* __restrict__ wc, const float* __restrict__ bcomb,
                             float* __restrict__ sval) {
  __shared__ float sd[64];
  int f = threadIdx.x;
  float g = gacc[f] / (float)NN + g2b[f];
  sd[f] = g * wc[64 + f];
  __syncthreads();
  for (int s = 32; s > 0; s >>= 1) { if (f < s) sd[f] += sd[f + s]; __syncthreads(); }
  if (f == 0) *sval = sd[0] + *bcomb;
}

__global__ void k_out(const float* __restrict__ H, const float* __restrict__ wc,
                      const float* __restrict__ sval, float* __restrict__ out) {
  int i = blockIdx.x * blockDim.x + threadIdx.x;
  if (i >= B_) return;
  const float4* row = (const float4*)(H + (size_t)i * F_);
  const float4* w4  = (const float4*)wc;
  float s = 0.f;
#pragma unroll
  for (int f = 0; f < 16; ++f) {
    float4 a = row[f], b = w4[f];
    s += a.x * b.x + a.y * b.y + a.z * b.z + a.w * b.w;
  }
  out[i] = s + *sval;
}

// ---------------------------------------------------------------------------
extern "C" void kernel_launch(void* const* d_in, const int* in_sizes, int n_in,
                              void* d_out, int out_size, void* d_ws, size_t ws_size,
                              hipStream_t stream) {
  const float*     x1   = (const float*)d_in[0];
  const float*     x2   = (const float*)d_in[1];
  const long long* ei   = (const long long*)d_in[2];
  const float*     dnnW = (const float*)d_in[3];
  const float*     dnnb = (const float*)d_in[4];
  const float*     gam  = (const float*)d_in[5];
  const float*     bet  = (const float*)d_in[6];
  const float*     g1W  = (const float*)d_in[7];
  const float*     g1b  = (const float*)d_in[8];
  const float*     g2W  = (const float*)d_in[9];
  const float*     g2b  = (const float*)d_in[10];
  const float*     fc1W = (const float*)d_in[11];
  const float*     fc1b = (const float*)d_in[12];
  const float*     fc2W = (const float*)d_in[13];
  const float*     fc2b = (const float*)d_in[14];
  float* out = (float*)d_out;

  float* ws   = (float*)d_ws;
  float* A    = ws;                       // [NN*64] h1 / h2
  float* AG   = ws + 3200000;             // [NN*64] agg / g
  float* deg  = ws + 6400000;             // [NN] deg -> dis (in place)
  float* H    = ws + 6450000;             // [B_*64] dnn h -> dnn_emb (in place)
  float* mu   = ws + 7498576;             // [64]
  float* istd = ws + 7498640;             // [64]
  float* gacc = ws + 7498704;             // [64]
  float* wc   = ws + 7498768;             // [128]
  float* bcmb = ws + 7498896;             // [1]
  float* sval = ws + 7498897;             // [1]

  const int T = 256;
  // 1) init: agg=0, deg=1 (self-loop), gacc=0
  k_init<<<(NN * F_ + T - 1) / T, T, 0, stream>>>(AG, deg, gacc);
  // 2) degree, 3) dis = rsqrt(deg)
  k_degree<<<(NE + T - 1) / T, T, 0, stream>>>(ei, deg);
  k_rsqrt_inplace<<<(NN + T - 1) / T, T, 0, stream>>>(deg, NN);
  // 4) h1 = x2 @ gcn1_W        (3125 row tiles, 8 waves/block)
  gemm_wmma_f32_n64<<<(NN / 16 + 7) / 8, T, 0, stream>>>(x2, g1W, nullptr, A, NN, F_);
  // 5) h = x1 @ dnn_W + dnn_b  (1024 row tiles)
  gemm_wmma_f32_n64<<<(B_ / 16 + 7) / 8, T, 0, stream>>>(x1, dnnW, dnnb, H, B_, KDNN);
  // 6) layer-1 scatter, 7) g = relu(agg + dis^2*h1 + b1) in place in AG
  k_aggregate<<<NE * 16 / T, T, 0, stream>>>(ei, deg, A, AG);
  k_combine_relu<<<(NN * F_ + T - 1) / T, T, 0, stream>>>(AG, A, deg, g1b);
  // 8) h2 = g @ gcn2_W  (overwrites A)
  gemm_wmma_f32_n64<<<(NN / 16 + 7) / 8, T, 0, stream>>>(AG, g2W, nullptr, A, NN, F_);
  // 9) re-zero agg, 10) layer-2 scatter, 11) column mean accumulation
  k_zero<<<(NN * F_ + T - 1) / T, T, 0, stream>>>(AG, NN * F_);
  k_aggregate<<<NE * 16 / T, T, 0, stream>>>(ei, deg, A, AG);
  k_combine_mean<<<256, T, 0, stream>>>(AG, A, deg, gacc);
  // 12/13) BatchNorm (batch stats) + ReLU, in place in H
  k_bnstats<<<F_, T, 0, stream>>>(H, mu, istd);
  k_bnapply_relu<<<(B_ * F_ + T - 1) / T, T, 0, stream>>>(H, mu, istd, gam, bet);
  // 14) fold fc1@fc2, 15) GNN scalar, 16) final per-row dot
  k_foldw<<<1, 128, 0, stream>>>(fc1W, fc1b, fc2W, fc2b, wc, bcmb);
  k_gnn_scalar<<<1, 64, 0, stream>>>(gacc, g2b, wc, bcmb, sval);
  k_out<<<(B_ + T - 1) / T, T, 0, stream>>>(H, wc, sval, out);
}